// GIN_77043123355994
// MI455X (gfx1250) — compile-verified
//
#include <hip/hip_runtime.h>
#include <hip/hip_bf16.h>

// ---------------------------------------------------------------------------
// GIN (3 layers, HIDDEN=128) for MI455X / gfx1250.
//  - fp32 WMMA (v_wmma_f32_16x16x4_f32) for both MLP GEMMs
//  - native global f32 atomics for the edge scatter (L2-resident: x is 25.6MB
//    vs 192MB L2, so gather+atomic traffic stays on-chip)
//  - LDS-staged GEMM tiles with bank-conflict-free strides
//  - BN stats fused into GEMM2 epilogue via ds_add_f32 + one global atomic
// ---------------------------------------------------------------------------

#define HIDDEN 128
#define GIN_LAYERS 3
#define BN_EPS 1e-5f

typedef __attribute__((ext_vector_type(2))) float v2f;
typedef __attribute__((ext_vector_type(8))) float v8f;

// LDS strides (in floats) chosen so WMMA fragment reads are conflict-free:
//  A read: lanes 0-15 rows m+0..15 same col -> banks c+4r (stride 132) distinct;
//          lanes 16-31 read col c+2 -> offset bank set, no overlap.
//  W read: lanes 0-15 row k cols n..n+15; lanes 16-31 row k+2 -> 2*136 % 64 = 16
//          so their banks are base+16..31, disjoint from base+0..15.
constexpr int A_STRIDE = 132;
constexpr int W_STRIDE = 136;

__device__ __forceinline__ void atomic_add_global_f32(float* p, float v) {
    __hip_atomic_fetch_add(p, v, __ATOMIC_RELAXED, __HIP_MEMORY_SCOPE_AGENT);
}
__device__ __forceinline__ void atomic_add_lds_f32(float* p, float v) {
    __hip_atomic_fetch_add(p, v, __ATOMIC_RELAXED, __HIP_MEMORY_SCOPE_WORKGROUP);
}

// agg = x   (vectorized copy; also the "+x" of GIN's (1+eps)*x with eps=0)
__global__ void gin_copy_f4(const float4* __restrict__ src,
                            float4* __restrict__ dst, int n4) {
    int i = blockIdx.x * blockDim.x + threadIdx.x;
    if (i < n4) dst[i] = src[i];
}

// agg[dst[e]][c] += x[src[e]][c]  -- one thread per (edge, channel)
__global__ void gin_scatter_add(const float* __restrict__ x,
                                const int* __restrict__ src,
                                const int* __restrict__ dst,
                                float* __restrict__ agg, int nEdges) {
    int idx = blockIdx.x * blockDim.x + threadIdx.x;
    int e = idx >> 7;          // HIDDEN == 128
    int c = idx & 127;
    if (e < nEdges) {
        int s = src[e];
        int d = dst[e];
        atomic_add_global_f32(&agg[(size_t)d * HIDDEN + c],
                              x[(size_t)s * HIDDEN + c]);
    }
}

__global__ void gin_zero_stats(float* __restrict__ g) {
    g[threadIdx.x] = 0.0f;     // 256 threads: sum[128] + sumsq[128]
}

// out = relu(A @ W + bias), A:[M,128], W:[128,128] row-major [k][n].
// 256 threads = 8 waves; block tile = 128 rows x 128 cols.
// Wave w computes rows [blk*128 + w*16, +16) for all 8 n-tiles of 16.
// WITH_STATS additionally accumulates per-column sum / sum^2 into gstats.
template <bool WITH_STATS>
__global__ __launch_bounds__(256)
void gin_mlp_gemm(const float* __restrict__ A, const float* __restrict__ W,
                  const float* __restrict__ bias, float* __restrict__ out,
                  float* __restrict__ gstats, int M) {
    extern __shared__ float smem[];
    float* As   = smem;                                  // 128 * A_STRIDE
    float* Ws   = smem + 128 * A_STRIDE;                 // 128 * W_STRIDE
    float* lsum = Ws + 128 * W_STRIDE;                   // 128
    float* lsq  = lsum + 128;                            // 128

    const int tid   = threadIdx.x;
    const int mBase = blockIdx.x * 128;

    if (WITH_STATS && tid < 128) { lsum[tid] = 0.0f; lsq[tid] = 0.0f; }

    // Stage A tile (zero-padded past M) and W, coalesced float4 loads.
#pragma unroll
    for (int it = 0; it < 16; ++it) {
        int i  = it * 256 + tid;        // 0..4095 float4 slots
        int r  = i >> 5;
        int c4 = (i & 31) << 2;
        float4 v = make_float4(0.f, 0.f, 0.f, 0.f);
        int row = mBase + r;
        if (row < M) v = *(const float4*)(A + (size_t)row * HIDDEN + c4);
        float* p = As + r * A_STRIDE + c4;
        p[0] = v.x; p[1] = v.y; p[2] = v.z; p[3] = v.w;
    }
#pragma unroll
    for (int it = 0; it < 16; ++it) {
        int i  = it * 256 + tid;
        int r  = i >> 5;
        int c4 = (i & 31) << 2;
        float4 v = *(const float4*)(W + (size_t)r * HIDDEN + c4);
        float* p = Ws + r * W_STRIDE + c4;
        p[0] = v.x; p[1] = v.y; p[2] = v.z; p[3] = v.w;
    }
    __syncthreads();

    const int wave  = tid >> 5;
    const int lane  = tid & 31;
    const int m0    = wave * 16;
    const int lrow  = lane & 15;    // A row / B-C-D column within tile
    const int khalf = lane >> 4;    // K-half select (ISA 16x4 f32 A layout)

    // Hoist this wave's A fragments for all K into registers (64 VGPRs).
    float areg[64];
#pragma unroll
    for (int kk = 0; kk < 32; ++kk) {
        int k = kk * 4 + khalf * 2;
        areg[2 * kk]     = As[(m0 + lrow) * A_STRIDE + k];
        areg[2 * kk + 1] = As[(m0 + lrow) * A_STRIDE + k + 1];
    }

    // Wave-uniform guard: EXEC stays all-1s inside (WMMA requirement).
    const bool rowsValid = (mBase + m0) < M;

#pragma unroll
    for (int nt = 0; nt < 8; ++nt) {
        v8f acc = {};
        if (rowsValid) {
#pragma unroll
            for (int kk = 0; kk < 32; ++kk) {
                int k = kk * 4 + khalf * 2;
                v2f a; a.x = areg[2 * kk]; a.y = areg[2 * kk + 1];
                v2f b;
                b.x = Ws[k * W_STRIDE + nt * 16 + lrow];
                b.y = Ws[(k + 1) * W_STRIDE + nt * 16 + lrow];
                acc = __builtin_amdgcn_wmma_f32_16x16x4_f32(
                    false, a, false, b, (short)0, acc, false, false);
            }
        }
        // Epilogue: + bias, relu, store, (stats).
        const int n  = nt * 16 + lrow;
        const float bv = bias[n];
        float ssum = 0.0f, ssq = 0.0f;
        const int rbase = mBase + m0 + khalf * 8;
#pragma unroll
        for (int v = 0; v < 8; ++v) {
            float o = acc[v] + bv;
            o = o > 0.0f ? o : 0.0f;
            int row = rbase + v;
            if (row < M) {
                out[(size_t)row * HIDDEN + n] = o;
                if (WITH_STATS) { ssum += o; ssq += o * o; }
            }
        }
        if (WITH_STATS) {
            atomic_add_lds_f32(&lsum[n], ssum);
            atomic_add_lds_f32(&lsq[n], ssq);
        }
    }

    if (WITH_STATS) {
        __syncthreads();
        if (tid < 128) {
            atomic_add_global_f32(&gstats[tid], lsum[tid]);
            atomic_add_global_f32(&gstats[128 + tid], lsq[tid]);
        }
    }
}

// stats[n] -> mean, stats[128+n] -> rsqrt(var + eps)   (128 threads)
__global__ void gin_bn_finalize(float* __restrict__ g, float inv_n) {
    int n = threadIdx.x;
    float mean = g[n] * inv_n;
    float var  = g[128 + n] * inv_n - mean * mean;
    g[n]       = mean;
    g[128 + n] = rsqrtf(var + BN_EPS);
}

// out = gamma * (h - mean) * invstd + beta   (float4 vectorized over columns)
__global__ void gin_bn_apply(const float* __restrict__ h,
                             const float* __restrict__ stats,
                             const float* __restrict__ gamma,
                             const float* __restrict__ beta,
                             float* __restrict__ out, int n4) {
    int i = blockIdx.x * blockDim.x + threadIdx.x;
    if (i >= n4) return;
    int c = (i & 31) << 2;   // column of first component
    float4 v = ((const float4*)h)[i];
    float4 o;
    o.x = gamma[c + 0] * (v.x - stats[c + 0]) * stats[128 + c + 0] + beta[c + 0];
    o.y = gamma[c + 1] * (v.y - stats[c + 1]) * stats[128 + c + 1] + beta[c + 1];
    o.z = gamma[c + 2] * (v.z - stats[c + 2]) * stats[128 + c + 2] + beta[c + 2];
    o.w = gamma[c + 3] * (v.w - stats[c + 3]) * stats[128 + c + 3] + beta[c + 3];
    ((float4*)out)[i] = o;
}

extern "C" void kernel_launch(void* const* d_in, const int* in_sizes, int n_in,
                              void* d_out, int out_size, void* d_ws, size_t ws_size,
                              hipStream_t stream) {
    const float* x0    = (const float*)d_in[0];
    const int*   ei    = (const int*)d_in[1];
    const float* W1    = (const float*)d_in[2];
    const float* b1    = (const float*)d_in[3];
    const float* W2    = (const float*)d_in[4];
    const float* b2    = (const float*)d_in[5];
    const float* gamma = (const float*)d_in[6];
    const float* beta  = (const float*)d_in[7];

    const int M = in_sizes[0] / HIDDEN;     // 50000 nodes
    const int E = in_sizes[1] / 2;          // 800000 edges
    const int* srcIdx = ei;
    const int* dstIdx = ei + E;
    float* out = (float*)d_out;

    const size_t NH = (size_t)M * HIDDEN;
    float* agg   = (float*)d_ws;            // [M,128] agg, reused as H
    float* tbuf  = agg + NH;                // [M,128] intermediate / next-x
    float* stats = tbuf + NH;               // [256] sum + sumsq -> mean + invstd

    const int n4 = (int)(NH / 4);
    const int copyBlocks    = (n4 + 255) / 256;
    const int scatterBlocks = (int)(((long long)E * HIDDEN + 255) / 256);
    const int gemmBlocks    = (M + 127) / 128;
    const size_t gemmLds =
        (size_t)(128 * A_STRIDE + 128 * W_STRIDE + 256) * sizeof(float);
    const float inv_n = 1.0f / (float)M;

    const float* xin = x0;
    for (int layer = 0; layer < GIN_LAYERS; ++layer) {
        const float* w1 = W1 + (size_t)layer * HIDDEN * HIDDEN;
        const float* w2 = W2 + (size_t)layer * HIDDEN * HIDDEN;
        const float* bb1 = b1 + (size_t)layer * HIDDEN;
        const float* bb2 = b2 + (size_t)layer * HIDDEN;

        // agg = x ; agg[dst] += x[src]
        gin_copy_f4<<<copyBlocks, 256, 0, stream>>>(
            (const float4*)xin, (float4*)agg, n4);
        gin_scatter_add<<<scatterBlocks, 256, 0, stream>>>(
            xin, srcIdx, dstIdx, agg, E);

        // t = relu(agg @ W1 + b1)
        gin_mlp_gemm<false><<<gemmBlocks, 256, gemmLds, stream>>>(
            agg, w1, bb1, tbuf, nullptr, M);

        // h = relu(t @ W2 + b2), with fused column sum/sumsq  (h overwrites agg)
        gin_zero_stats<<<1, 256, 0, stream>>>(stats);
        gin_mlp_gemm<true><<<gemmBlocks, 256, gemmLds, stream>>>(
            tbuf, w2, bb2, agg, stats, M);

        // BN: finalize stats, then normalize. Last layer writes d_out,
        // otherwise writes tbuf (becomes next layer's x).
        gin_bn_finalize<<<1, 128, 0, stream>>>(stats, inv_n);
        float* lout = (layer == GIN_LAYERS - 1) ? out : tbuf;
        gin_bn_apply<<<copyBlocks, 256, 0, stream>>>(
            agg, stats, gamma, beta, lout, n4);

        xin = tbuf;
    }
}